// Detection_28948079575399
// MI455X (gfx1250) — compile-verified
//
#include <hip/hip_runtime.h>
#include <cstdint>

// ---------------- problem constants (from reference) ----------------
constexpr int   N_   = 8;
constexpr int   A_   = 16384;
constexpr int   C_   = 21;
constexpr int   CM1_ = 20;
constexpr int   K_   = 100;
constexpr float PROB_THR = 0.05f;
constexpr float NMS_THR  = 0.45f;

// ---------------- CDNA5 TDM (tensor_load_to_lds) ----------------
typedef unsigned int u32x4 __attribute__((ext_vector_type(4)));
typedef int          i32x8 __attribute__((ext_vector_type(8)));
typedef int          i32x4 __attribute__((ext_vector_type(4)));

// Build a D# for a contiguous row of `elems` f32 values and issue the TDM load.
// Group0: count=1, lds_addr, global_addr[56:0], type=2.
// Group1: data_size=4B, tensor_dim0=elems, tensor_dim1=1, tile_dim0=elems,
//         tile_dim1=1, tensor_dim0_stride=elems, workgroup_mask=0 (no cluster).
__device__ __forceinline__ void tdm_load_row_f32(const float* gsrc,
                                                 unsigned lds_byte_off,
                                                 int elems) {
  unsigned long long ga = (unsigned long long)(uintptr_t)gsrc;
  u32x4 g0;
  g0[0] = 1u;                                               // count=1 (user D#)
  g0[1] = lds_byte_off;                                     // lds_addr
  g0[2] = (unsigned)(ga & 0xFFFFFFFFull);                   // global_addr lo
  g0[3] = ((unsigned)(ga >> 32) & 0x01FFFFFFu) | (2u << 30); // addr hi | type=2
  i32x8 g1;
  g1[0] = 0x00020000;                                        // data_size=2 (4B)
  g1[1] = (int)(((unsigned)elems & 0xFFFFu) << 16);          // tensor_dim0 lo16
  g1[2] = (int)((((unsigned)elems >> 16) & 0xFFFFu) | (1u << 16)); // dim0 hi | dim1 lo
  g1[3] = (int)(((unsigned)elems & 0xFFFFu) << 16);          // dim1 hi(0) | tile_dim0
  g1[4] = 1;                                                 // tile_dim1=1, tile_dim2=0
  g1[5] = elems;                                             // tensor_dim0_stride lo
  g1[6] = 0;
  g1[7] = 0;
  i32x4 z4 = {0, 0, 0, 0};                                   // 2-D: groups 2/3 unused
  i32x8 z8 = {0, 0, 0, 0, 0, 0, 0, 0};
  __builtin_amdgcn_tensor_load_to_lds(g0, g1, z4, z4, z8, 0);
}

// ---------------- kernel 1: softmax + threshold + box decode ----------------
__global__ __launch_bounds__(256) void k_prep(const float* __restrict__ cls,
                                              const float* __restrict__ reg,
                                              const float* __restrict__ anc,
                                              float* __restrict__ boxes,
                                              float* __restrict__ scores) {
  int gid = blockIdx.x * 256 + threadIdx.x;       // n*A + a
  if (gid >= N_ * A_) return;
  int n = gid / A_;
  int a = gid - n * A_;

  const float* cp = cls + (size_t)gid * C_;
  float v[C_];
  float m = -1e30f;
#pragma unroll
  for (int i = 0; i < C_; ++i) { v[i] = cp[i]; m = fmaxf(m, v[i]); }
  float s = 0.0f;
#pragma unroll
  for (int i = 0; i < C_; ++i) { v[i] = __expf(v[i] - m); s += v[i]; }
  float inv = 1.0f / s;
#pragma unroll
  for (int i = 1; i < C_; ++i) {
    float p = v[i] * inv;
    scores[((size_t)n * CM1_ + (i - 1)) * A_ + a] = (p > PROB_THR) ? p : -1.0f;
  }

  const float4 l  = ((const float4*)reg)[gid];
  const float4 an = ((const float4*)anc)[a];
  float cx = an.x + l.x * 0.1f * an.z;
  float cy = an.y + l.y * 0.1f * an.w;
  float w  = an.z * __expf(l.z * 0.2f);
  float h  = an.w * __expf(l.w * 0.2f);
  float4 b;
  b.x = fminf(fmaxf(cx - 0.5f * w, 0.0f), 1.0f);
  b.y = fminf(fmaxf(cy - 0.5f * h, 0.0f), 1.0f);
  b.z = fminf(fmaxf(cx + 0.5f * w, 0.0f), 1.0f);
  b.w = fminf(fmaxf(cy + 0.5f * h, 0.0f), 1.0f);
  ((float4*)boxes)[gid] = b;
}

// ---------------- kernel 2: per-(image,class) sequential NMS ----------------
__global__ __launch_bounds__(1024) void k_nms(const float* __restrict__ boxes,
                                              const float* __restrict__ scores,
                                              float* __restrict__ rows) {
  const int n = blockIdx.x / C_;
  const int c = blockIdx.x % C_;
  const int t = threadIdx.x;

  float* rowOut = rows + ((size_t)n * C_ + c) * (K_ * 6);
  for (int j = t; j < K_ * 6; j += 1024) rowOut[j] = 0.0f;  // rows default to zero
  if (c == 0) return;                                        // background class

  __shared__ float sLds[A_];        // 64 KB score plane staged via TDM
  __shared__ float redv[32];
  __shared__ int   redi[32];
  __shared__ float bcast[6];        // [x0,y0,x1,y1,score]
  __shared__ int   bidx;

  const float* sp = scores + ((size_t)n * CM1_ + (c - 1)) * A_;
  if (t < 32) {                     // wave 0 issues the async tensor DMA
    unsigned off = (unsigned)(uintptr_t)(&sLds[0]);
    tdm_load_row_f32(sp, off, A_);
    __builtin_amdgcn_s_wait_tensorcnt(0);
  }
  __syncthreads();

  constexpr int PT = A_ / 1024;     // 16 anchors/thread, strided for coalescing
  float  sv[PT], ar[PT];
  float4 bx[PT];
  const float4* bp = (const float4*)boxes + (size_t)n * A_;
#pragma unroll
  for (int i = 0; i < PT; ++i) {
    int a = t + i * 1024;
    sv[i] = sLds[a];
    bx[i] = bp[a];
    ar[i] = (bx[i].z - bx[i].x) * (bx[i].w - bx[i].y);
  }

  for (int it = 0; it < K_; ++it) {
    // per-thread argmax (indices ascending -> strict '>' keeps lowest index)
    float bv = sv[0]; int bi = t;
#pragma unroll
    for (int i = 1; i < PT; ++i)
      if (sv[i] > bv) { bv = sv[i]; bi = t + i * 1024; }
    // wave32 shuffle reduction with lowest-index tie-break (jnp.argmax semantics)
#pragma unroll
    for (int m = 16; m >= 1; m >>= 1) {
      float ov = __shfl_xor(bv, m, 32);
      int   oi = __shfl_xor(bi, m, 32);
      if (ov > bv || (ov == bv && oi < bi)) { bv = ov; bi = oi; }
    }
    if ((t & 31) == 0) { redv[t >> 5] = bv; redi[t >> 5] = bi; }
    __syncthreads();
    if (t < 32) {                    // wave 0 reduces the 32 wave results
      bv = redv[t]; bi = redi[t];
#pragma unroll
      for (int m = 16; m >= 1; m >>= 1) {
        float ov = __shfl_xor(bv, m, 32);
        int   oi = __shfl_xor(bi, m, 32);
        if (ov > bv || (ov == bv && oi < bi)) { bv = ov; bi = oi; }
      }
      if (t == 0) {
        bcast[4] = bv; bidx = bi;
        if (bv > 0.0f) {
          float4 sb = bp[bi];        // L2-resident single box fetch
          bcast[0] = sb.x; bcast[1] = sb.y; bcast[2] = sb.z; bcast[3] = sb.w;
        }
      }
    }
    __syncthreads();

    float bestv = bcast[4];
    if (bestv <= 0.0f) break;        // uniform: nothing can become valid later
    int   besti = bidx;
    float sx0 = bcast[0], sy0 = bcast[1], sx1 = bcast[2], sy1 = bcast[3];
    float areaS = (sx1 - sx0) * (sy1 - sy0);

#pragma unroll
    for (int i = 0; i < PT; ++i) {   // register-resident IoU suppression sweep
      int a = t + i * 1024;
      float lx = fmaxf(sx0, bx[i].x);
      float ly = fmaxf(sy0, bx[i].y);
      float rx = fminf(sx1, bx[i].z);
      float ry = fminf(sy1, bx[i].w);
      float iw = fmaxf(rx - lx, 0.0f);
      float ih = fmaxf(ry - ly, 0.0f);
      float inter = iw * ih;
      float uni = areaS + ar[i] - inter;
      float iou = (uni > 0.0f) ? (inter / uni) : 0.0f;
      if (iou > NMS_THR || a == besti) sv[i] = -1.0f;
    }
    if (t == 0) {
      rowOut[it * 6 + 0] = sx0;
      rowOut[it * 6 + 1] = sy0;
      rowOut[it * 6 + 2] = sx1;
      rowOut[it * 6 + 3] = sy1;
      rowOut[it * 6 + 4] = bestv;
      rowOut[it * 6 + 5] = (float)c;
    }
  }
}

// ---------------- kernel 3: per-image stable top-100 of 2100 rows ----------------
__global__ __launch_bounds__(256) void k_topk(const float* __restrict__ rows,
                                              float* __restrict__ out) {
  const int n = blockIdx.x;
  const int t = threadIdx.x;
  constexpr int M = C_ * K_;        // 2100
  __shared__ float vals[M];
  __shared__ float rv[8];
  __shared__ int   ri[8];

  const float* r = rows + (size_t)n * M * 6;
  for (int j = t; j < M; j += 256) vals[j] = r[(size_t)j * 6 + 4];
  __syncthreads();

  float* o = out + (size_t)n * K_ * 6;
  for (int k = 0; k < K_; ++k) {
    float bv = -1e30f; int bi = 0x7FFFFFFF;
    for (int j = t; j < M; j += 256) {
      float v = vals[j];
      if (v > bv) { bv = v; bi = j; }  // j ascending -> lowest index on ties
    }
#pragma unroll
    for (int m = 16; m >= 1; m >>= 1) {
      float ov = __shfl_xor(bv, m, 32);
      int   oi = __shfl_xor(bi, m, 32);
      if (ov > bv || (ov == bv && oi < bi)) { bv = ov; bi = oi; }
    }
    if ((t & 31) == 0) { rv[t >> 5] = bv; ri[t >> 5] = bi; }
    __syncthreads();
    if (t == 0) {
      float fbv = rv[0]; int fbi = ri[0];
      for (int w = 1; w < 8; ++w)
        if (rv[w] > fbv || (rv[w] == fbv && ri[w] < fbi)) { fbv = rv[w]; fbi = ri[w]; }
      vals[fbi] = -2.0f;            // remove (all live scores are >= 0)
      const float* src = r + (size_t)fbi * 6;
#pragma unroll
      for (int q = 0; q < 6; ++q) o[k * 6 + q] = src[q];
    }
    __syncthreads();
  }
}

// ---------------- host launch ----------------
extern "C" void kernel_launch(void* const* d_in, const int* in_sizes, int n_in,
                              void* d_out, int out_size, void* d_ws, size_t ws_size,
                              hipStream_t stream) {
  const float* cls = (const float*)d_in[0];   // [8,16384,21]
  const float* reg = (const float*)d_in[1];   // [8,16384,4]
  const float* anc = (const float*)d_in[2];   // [16384,4]
  float* ws     = (float*)d_ws;
  float* boxes  = ws;                                   // N*A*4      (2.0 MB)
  float* scores = boxes + (size_t)N_ * A_ * 4;          // N*20*A     (10.5 MB)
  float* rows   = scores + (size_t)N_ * CM1_ * A_;      // N*21*100*6 (0.4 MB)
  float* outp   = (float*)d_out;                        // [8,100,6]

  k_prep<<<(N_ * A_ + 255) / 256, 256, 0, stream>>>(cls, reg, anc, boxes, scores);
  k_nms<<<N_ * C_, 1024, 0, stream>>>(boxes, scores, rows);
  k_topk<<<N_, 256, 0, stream>>>(rows, outp);
}